// ChebyDecimate_32255204393780
// MI455X (gfx1250) — compile-verified
//
#include <hip/hip_runtime.h>
#include <math.h>

#ifndef M_PI
#define M_PI 3.14159265358979323846
#endif

typedef __attribute__((ext_vector_type(2))) float v2f;
typedef __attribute__((ext_vector_type(8))) float v8f;

#define TLEN   131072
#define BATCH  32
#define LCH    128                 // chunk length (samples)
#define PCHUNK (TLEN / LCH)        // 1024 chunks per row
#define ROWS   (PCHUNK + 16)       // +16 zero guard slots per row
#define NCH    (BATCH * PCHUNK)    // 32768 chunks total -> 1024 wave32 waves

// workspace layout in floats (total ~529K floats ~= 2.1 MB)
#define WS_COEF  0                          // 20 floats: (b0,b1,b2,a1,a2) x 4 sections
#define WS_PHI   32                         // 3 x 256 floats: Phi^1..Phi^3, 16x16 zero-padded, row-major
#define WS_V     1024                       // BATCH*ROWS*8 chunk zero-state final states
#define WS_SINIT (WS_V + BATCH * ROWS * 8)  // NCH*8 corrected initial states

// --------------------------------------------------------------------------
// Setup: derive SOS coefficients (double, mirrors scipy cheby1 math) and the
// chunk transition matrices Phi^k = (A^LCH)^k by probing the homogeneous
// recurrence from the 8 basis states. Single thread; negligible cost.
// --------------------------------------------------------------------------
__global__ void ChebySetup_k(float* __restrict__ ws) {
  if (threadIdx.x != 0 || blockIdx.x != 0) return;
  const int N = 8;
  const double rp = 0.05, Wn = 0.5;
  double eps = sqrt(pow(10.0, rp / 10.0) - 1.0);
  double mu  = asinh(1.0 / eps) / (double)N;
  double shm = sinh(mu), chm = cosh(mu);
  double pr[8], pim[8];
  for (int k = 0; k < N; ++k) {
    double th = M_PI * (2.0 * (k + 1) - 1.0) / (2.0 * N);
    pr[k]  = -shm * sin(th);
    pim[k] =  chm * cos(th);
  }
  // gain = Re(prod(-p)) / sqrt(1+eps^2)   (N even)
  double gr = 1.0, gi = 0.0;
  for (int k = 0; k < N; ++k) {
    double ar = -pr[k], ai = -pim[k];
    double nr = gr * ar - gi * ai;
    gi = gr * ai + gi * ar;
    gr = nr;
  }
  double gain = gr / sqrt(1.0 + eps * eps);
  const double fs = 2.0;
  double warped = 2.0 * fs * tan(M_PI * Wn / fs);   // = 4 for Wn = 0.5
  for (int k = 0; k < N; ++k) { pr[k] *= warped; pim[k] *= warped; }
  gain *= pow(warped, (double)N);
  const double fs2 = 2.0 * fs;
  double dr = 1.0, di = 0.0;
  double zr[8], zim[8];
  for (int k = 0; k < N; ++k) {
    double ar = fs2 - pr[k], ai = -pim[k];   // (fs2 - p)
    double nr = dr * ar - di * ai;
    di = dr * ai + di * ar;
    dr = nr;
    double numr = fs2 + pr[k], numi = pim[k];
    double den  = ar * ar + ai * ai;
    zr[k]  = (numr * ar + numi * ai) / den;  // pz = (fs2+p)/(fs2-p)
    zim[k] = (numi * ar - numr * ai) / den;
  }
  gain /= dr;
  double g = pow(gain, 0.25);                // gain spread over 4 sections
  double db0[4], db1[4], db2[4], da1[4], da2[4];
  int s = 0;
  for (int k = 0; k < N; ++k) {
    if (zim[k] > 0.0 && s < 4) {
      da1[s] = -2.0 * zr[k];
      da2[s] = zr[k] * zr[k] + zim[k] * zim[k];
      db0[s] = g; db1[s] = 2.0 * g; db2[s] = g;
      ++s;
    }
  }
  for (int q = 0; q < 4; ++q) {
    ws[WS_COEF + q * 5 + 0] = (float)db0[q];
    ws[WS_COEF + q * 5 + 1] = (float)db1[q];
    ws[WS_COEF + q * 5 + 2] = (float)db2[q];
    ws[WS_COEF + q * 5 + 3] = (float)da1[q];
    ws[WS_COEF + q * 5 + 4] = (float)da2[q];
  }
  // Phi = A^LCH by basis probing (state = [z1_0,z2_0,z1_1,z2_1,...])
  double P1[8][8], P2[8][8], P3[8][8];
  for (int j = 0; j < 8; ++j) {
    double z[8];
    for (int i = 0; i < 8; ++i) z[i] = 0.0;
    z[j] = 1.0;
    for (int t = 0; t < LCH; ++t) {
      double y = 0.0;                         // homogeneous: x = 0
      for (int q = 0; q < 4; ++q) {
        double out = db0[q] * y + z[2 * q];
        double nz1 = db1[q] * y + z[2 * q + 1] - da1[q] * out;
        double nz2 = db2[q] * y - da2[q] * out;
        z[2 * q] = nz1; z[2 * q + 1] = nz2;
        y = out;
      }
    }
    for (int i = 0; i < 8; ++i) P1[i][j] = z[i];
  }
  for (int i = 0; i < 8; ++i)
    for (int j = 0; j < 8; ++j) {
      double acc = 0.0;
      for (int k = 0; k < 8; ++k) acc += P1[i][k] * P1[k][j];
      P2[i][j] = acc;
    }
  for (int i = 0; i < 8; ++i)
    for (int j = 0; j < 8; ++j) {
      double acc = 0.0;
      for (int k = 0; k < 8; ++k) acc += P2[i][k] * P1[k][j];
      P3[i][j] = acc;
    }
  for (int m = 0; m < 3; ++m)
    for (int r = 0; r < 16; ++r)
      for (int c = 0; c < 16; ++c) {
        double v = 0.0;
        if (r < 8 && c < 8) v = (m == 0) ? P1[r][c] : (m == 1 ? P2[r][c] : P3[r][c]);
        ws[WS_PHI + m * 256 + r * 16 + c] = (float)v;
      }
}

// one DF2T cascade step (fp32, matches reference structure)
__device__ __forceinline__ float cheby_step(float xin, float z1[4], float z2[4],
                                            const float b0[4], const float b1[4],
                                            const float b2[4], const float a1[4],
                                            const float a2[4]) {
  float y = xin;
#pragma unroll
  for (int s = 0; s < 4; ++s) {
    float out = fmaf(b0[s], y, z1[s]);
    z1[s] = fmaf(b1[s], y, z2[s]) - a1[s] * out;
    z2[s] = fmaf(b2[s], y, -(a2[s] * out));
    y = out;
  }
  return y;
}

// --------------------------------------------------------------------------
// Pass A: zero-state response per chunk; record final 8-float state only.
// One lane per chunk; a wave covers 32 consecutive chunks = 16KB contiguous.
// --------------------------------------------------------------------------
__global__ __launch_bounds__(256) void ChebyPassA_k(const float* __restrict__ x,
                                                    float* __restrict__ ws) {
  const int tid = blockIdx.x * 256 + threadIdx.x;
  float* v_arr = ws + WS_V;
  // zero the 16 guard state-slots at the head of each row
  if (tid < BATCH * 16) {
    const int b = tid >> 4, slot = tid & 15;
    float* gz = v_arr + ((size_t)b * ROWS + slot) * 8;
#pragma unroll
    for (int i = 0; i < 8; ++i) gz[i] = 0.0f;
  }
  float b0[4], b1[4], b2[4], a1[4], a2[4];
#pragma unroll
  for (int s = 0; s < 4; ++s) {
    b0[s] = ws[WS_COEF + s * 5 + 0];
    b1[s] = ws[WS_COEF + s * 5 + 1];
    b2[s] = ws[WS_COEF + s * 5 + 2];
    a1[s] = ws[WS_COEF + s * 5 + 3];
    a2[s] = ws[WS_COEF + s * 5 + 4];
  }
  const int b = tid / PCHUNK, p = tid % PCHUNK;
  float z1[4] = {0.f, 0.f, 0.f, 0.f}, z2[4] = {0.f, 0.f, 0.f, 0.f};
  const float4* xp = (const float4*)(x + (size_t)b * TLEN + (size_t)p * LCH);
#pragma unroll 4
  for (int it = 0; it < LCH / 4; ++it) {
    __builtin_prefetch(xp + it + 8, 0, 0);   // global_prefetch_b8
    float4 v = xp[it];
    cheby_step(v.x, z1, z2, b0, b1, b2, a1, a2);
    cheby_step(v.y, z1, z2, b0, b1, b2, a1, a2);
    cheby_step(v.z, z1, z2, b0, b1, b2, a1, a2);
    cheby_step(v.w, z1, z2, b0, b1, b2, a1, a2);
  }
  float* vo = v_arr + ((size_t)b * ROWS + 16 + p) * 8;
  float4 lo = make_float4(z1[0], z2[0], z1[1], z2[1]);
  float4 hi = make_float4(z1[2], z2[2], z1[3], z2[3]);
  *(float4*)(vo)     = lo;
  *(float4*)(vo + 4) = hi;
}

// --------------------------------------------------------------------------
// Pass B: corrected initial state via truncated Neumann series
//   s_init[c] = v[c-1] + Phi v[c-2] + Phi^2 v[c-3] + Phi^3 v[c-4]
// (||Phi^4|| ~ 4e-16 << fp32 eps -> exact in fp32).
// Batched over 16 chunks per wave with v_wmma_f32_16x16x4_f32:
//   D(16x16) += Phi^k(16x16, zero-padded 8x8) * V_{k+1}(16x16)
// A layout (ISA 7.12.2): lane m=lane%16; VGPR0/1 hold K = 2*(lane/16)+{0,1}
// B layout (mirror):     lane n=lane%16; VGPR0/1 hold K = 2*(lane/16)+{0,1}
// C/D layout: VGPR g holds M=g (lanes 0-15) / M=g+8 (lanes 16-31), N=lane%16.
// Only Phi columns 0..7 are nonzero -> 2 K-steps per product, 6 WMMAs total.
// --------------------------------------------------------------------------
__global__ __launch_bounds__(256) void ChebyPassB_k(float* __restrict__ ws) {
  const int gtid = blockIdx.x * 256 + threadIdx.x;
  const int wave = gtid >> 5;
  const int lane = gtid & 31;
  const int n    = lane & 15;
  const int half = lane >> 4;
  const float* phi    = ws + WS_PHI;
  const float* v_arr  = ws + WS_V;
  float*       s_init = ws + WS_SINIT;
  const int brow = wave >> 6;              // PCHUNK/16 = 64 tiles per row
  const int tile = wave & 63;
  const size_t base = (size_t)brow * ROWS + 16 + (size_t)tile * 16;

  // C init = identity term v[c-1]; rows 8..15 (lanes 16..31) are padding -> 0
  v8f d;
  if (lane < 16) {
    d = *(const v8f*)(v_arr + (base + (size_t)n - 1) * 8);
  } else {
#pragma unroll
    for (int g = 0; g < 8; ++g) d[g] = 0.0f;
  }
#pragma unroll
  for (int k = 1; k <= 3; ++k) {
    const float* M  = phi + (k - 1) * 256;                      // Phi^k padded 16x16
    const float* vv = v_arr + (base + (size_t)n - (size_t)(k + 1)) * 8;
#pragma unroll
    for (int kb = 0; kb < 2; ++kb) {
      const int kcol = 4 * kb + 2 * half;
      v2f a, bt;
      a.x  = M[n * 16 + kcol];
      a.y  = M[n * 16 + kcol + 1];
      bt.x = vv[kcol];
      bt.y = vv[kcol + 1];
      d = __builtin_amdgcn_wmma_f32_16x16x4_f32(false, a, false, bt, (short)0, d,
                                                false, false);
    }
  }
  if (lane < 16) {
    *(v8f*)(s_init + ((size_t)brow * PCHUNK + (size_t)tile * 16 + n) * 8) = d;
  }
}

// --------------------------------------------------------------------------
// Pass C: rerun each chunk from corrected state; emit decimated (even t)
// outputs packed as float4 (x re-read hits L2: 16.8MB << 192MB).
// --------------------------------------------------------------------------
__global__ __launch_bounds__(256) void ChebyPassC_k(const float* __restrict__ x,
                                                    const float* __restrict__ ws,
                                                    float* __restrict__ y) {
  const int tid = blockIdx.x * 256 + threadIdx.x;
  float b0[4], b1[4], b2[4], a1[4], a2[4];
#pragma unroll
  for (int s = 0; s < 4; ++s) {
    b0[s] = ws[WS_COEF + s * 5 + 0];
    b1[s] = ws[WS_COEF + s * 5 + 1];
    b2[s] = ws[WS_COEF + s * 5 + 2];
    a1[s] = ws[WS_COEF + s * 5 + 3];
    a2[s] = ws[WS_COEF + s * 5 + 4];
  }
  const float* s_init = ws + WS_SINIT;
  const int b = tid / PCHUNK, p = tid % PCHUNK;
  const float* sp = s_init + (size_t)tid * 8;
  float z1[4], z2[4];
#pragma unroll
  for (int s = 0; s < 4; ++s) { z1[s] = sp[2 * s]; z2[s] = sp[2 * s + 1]; }
  const float4* xp = (const float4*)(x + (size_t)b * TLEN + (size_t)p * LCH);
  float4* yp = (float4*)(y + (size_t)b * (TLEN / 2) + (size_t)p * (LCH / 2));
#pragma unroll 2
  for (int it = 0; it < LCH / 8; ++it) {
    __builtin_prefetch(xp + 2 * it + 8, 0, 0);
    float4 va = xp[2 * it], vb = xp[2 * it + 1];
    float4 o;
    o.x = cheby_step(va.x, z1, z2, b0, b1, b2, a1, a2);
    cheby_step(va.y, z1, z2, b0, b1, b2, a1, a2);
    o.y = cheby_step(va.z, z1, z2, b0, b1, b2, a1, a2);
    cheby_step(va.w, z1, z2, b0, b1, b2, a1, a2);
    o.z = cheby_step(vb.x, z1, z2, b0, b1, b2, a1, a2);
    cheby_step(vb.y, z1, z2, b0, b1, b2, a1, a2);
    o.w = cheby_step(vb.z, z1, z2, b0, b1, b2, a1, a2);
    cheby_step(vb.w, z1, z2, b0, b1, b2, a1, a2);
    yp[it] = o;
  }
}

extern "C" void kernel_launch(void* const* d_in, const int* in_sizes, int n_in,
                              void* d_out, int out_size, void* d_ws, size_t ws_size,
                              hipStream_t stream) {
  (void)in_sizes; (void)n_in; (void)out_size; (void)ws_size;
  const float* x = (const float*)d_in[0];
  float* out = (float*)d_out;
  float* ws  = (float*)d_ws;   // needs ~2.1 MB
  ChebySetup_k<<<1, 32, 0, stream>>>(ws);
  ChebyPassA_k<<<NCH / 256, 256, 0, stream>>>(x, ws);
  ChebyPassB_k<<<(NCH / 16) * 32 / 256, 256, 0, stream>>>(ws);
  ChebyPassC_k<<<NCH / 256, 256, 0, stream>>>(x, ws, out);
}